// FuzzyRGCNLayer_86053964742974
// MI455X (gfx1250) — compile-verified
//
#include <hip/hip_runtime.h>

typedef __attribute__((ext_vector_type(2))) float v2f;
typedef __attribute__((ext_vector_type(4))) float v4f;
typedef __attribute__((ext_vector_type(8))) float v8f;

#define NRELS  8
#define NRULES 4
#define INF    32
#define OUTF   32
#define KDIM   (NRULES * INF)   // 128 = folded (rule,in_feat) contraction dim
#define TILE   16               // edges per WMMA M-tile

// ---------------- pass 1: zero output + histogram counters ----------------
__global__ void k_init(float* __restrict__ out, int n, int* __restrict__ counts) {
  int t = blockIdx.x * blockDim.x + threadIdx.x;
  if (t < n) out[t] = 0.0f;
  if (blockIdx.x == 0 && t < NRELS) counts[t] = 0;
}

// ---------------- pass 2: histogram edges by relation ----------------------
__global__ void k_hist(const int* __restrict__ et, int* __restrict__ counts, int E) {
  int t = blockIdx.x * blockDim.x + threadIdx.x;
  if (t < E) atomicAdd(&counts[et[t]], 1);
}

// ---------------- pass 3: exclusive scan over 8 bins -----------------------
__global__ void k_scan(const int* __restrict__ counts, int* __restrict__ cursor) {
  if (blockIdx.x == 0 && threadIdx.x == 0) {
    int acc = 0;
    for (int r = 0; r < NRELS; ++r) { cursor[r] = acc; acc += counts[r]; }
  }
}

// ---------------- pass 4: bucket edges into relation-sorted order ----------
__global__ void k_scatter(const int* __restrict__ et, int* __restrict__ cursor,
                          int* __restrict__ order, int E) {
  int t = blockIdx.x * blockDim.x + threadIdx.x;
  if (t < E) {
    int pos = atomicAdd(&cursor[et[t]], 1);
    order[pos] = t;
  }
}

// ---------------- pass 5: WMMA GEMM tiles + bias + scatter-add -------------
// One wave32 per tile of 16 relation-sorted edges.
//   A (16 x 128) = tv[e,k] * feat[src[e], i]   (built on the fly from LDS)
//   B (128 x 32) = weight[r] viewed as (k*32+i, o), staged in LDS
//   D (16 x 32)  = msg, accumulated over 32 steps of v_wmma_f32_16x16x4_f32
__global__ __launch_bounds__(32) void k_rgcn_wmma(
    const float* __restrict__ feat,
    const float* __restrict__ tv,
    const float* __restrict__ weight,
    const float* __restrict__ hbias,
    const int*   __restrict__ etypes,
    const int*   __restrict__ src,
    const int*   __restrict__ dst,
    const int*   __restrict__ order,
    float*       __restrict__ out,
    int E)
{
  __shared__ float s_f[TILE][INF];      // 2 KB   staged source features
  __shared__ float s_w[KDIM * OUTF];    // 16 KB  staged relation weight (128x32)
  __shared__ float s_tv[TILE][NRULES];
  __shared__ int   s_src[TILE];
  __shared__ int   s_dst[TILE];
  __shared__ int   s_et[TILE];
  __shared__ int   s_relmask;

  const int lane = threadIdx.x;
  const int col  = lane & 15;   // N index (and A-row index for this lane)
  const int h    = lane >> 4;   // K-half / M-half selector
  const int base = blockIdx.x * TILE;

  if (lane == 0) s_relmask = 0;
  __syncthreads();

  // ---- load tile metadata (relation-sorted edge ids) ----
  if (lane < TILE) {
    int gi = base + lane;
    int e  = (gi < E) ? order[gi] : -1;
    int r = -1, s = 0, d = 0;
    float t0 = 0.f, t1 = 0.f, t2 = 0.f, t3 = 0.f;
    if (e >= 0) {
      r  = etypes[e]; s = src[e]; d = dst[e];
      t0 = tv[e * NRULES + 0]; t1 = tv[e * NRULES + 1];
      t2 = tv[e * NRULES + 2]; t3 = tv[e * NRULES + 3];
      atomicOr(&s_relmask, 1 << r);
    }
    s_et[lane] = r; s_src[lane] = s; s_dst[lane] = d;
    s_tv[lane][0] = t0; s_tv[lane][1] = t1;
    s_tv[lane][2] = t2; s_tv[lane][3] = t3;
  }
  __syncthreads();

  // ---- stage 16 gathered feature rows (coalesced: 32 lanes = 32 feats) ----
  #pragma unroll
  for (int j = 0; j < TILE; ++j)
    s_f[j][lane] = feat[s_src[j] * INF + lane];

  const int   relmask = s_relmask;
  const float tva = s_tv[col][0], tvb = s_tv[col][1];
  const float tvc = s_tv[col][2], tvd = s_tv[col][3];
  const int   myet = s_et[col];

  v8f acc0 = {};   // N = 0..15
  v8f acc1 = {};   // N = 16..31

  #pragma unroll 1
  for (int r = 0; r < NRELS; ++r) {
    if (!((relmask >> r) & 1)) continue;   // uniform branch; EXEC stays full
    __syncthreads();                       // prior pass done reading s_w
    // stage weight[r] -> LDS, 512B per b128 instruction across the wave
    const float* wr = weight + r * (KDIM * OUTF);
    #pragma unroll
    for (int j = 0; j < (KDIM * OUTF) / (32 * 4); ++j) {
      int idx = (j * 32 + lane) * 4;
      *(v4f*)(s_w + idx) = *(const v4f*)(wr + idx);
    }
    __syncthreads();

    // rows not of this relation contribute zero (data mask, not EXEC mask)
    const float msk  = (myet == r) ? 1.0f : 0.0f;
    const float mt[NRULES] = { tva * msk, tvb * msk, tvc * msk, tvd * msk };

    #pragma unroll
    for (int kk = 0; kk < KDIM / 4; ++kk) {      // 32 K-steps of 4
      const int   c0 = 4 * kk + 2 * h;           // this lane's K pair
      const int   i0 = c0 & (INF - 1);
      const float tk = mt[kk >> 3];              // rule index = c0/32 = kk/8
      v2f fp = *(const v2f*)&s_f[col][i0];       // ds_load_b64 (i0 even)
      v2f a;  a.x = tk * fp.x;  a.y = tk * fp.y;
      v2f b0; b0.x = s_w[ c0      * OUTF + col];
              b0.y = s_w[(c0 + 1) * OUTF + col];
      v2f b1; b1.x = s_w[ c0      * OUTF + 16 + col];
              b1.y = s_w[(c0 + 1) * OUTF + 16 + col];
      acc0 = __builtin_amdgcn_wmma_f32_16x16x4_f32(false, a, false, b0,
                                                   (short)0, acc0, false, false);
      acc1 = __builtin_amdgcn_wmma_f32_16x16x4_f32(false, a, false, b1,
                                                   (short)0, acc1, false, false);
    }
  }

  // ---- epilogue: add fuzzy-mixed bias, scatter-add to destination nodes ---
  #pragma unroll
  for (int v = 0; v < 8; ++v) {
    const int m = v + 8 * h;              // edge row in tile (C/D layout)
    if (base + m < E) {
      const int    d  = s_dst[m];
      const int    rr = s_et[m];
      const float  u0 = s_tv[m][0], u1 = s_tv[m][1];
      const float  u2 = s_tv[m][2], u3 = s_tv[m][3];
      const float* bb = hbias + rr * (NRULES * OUTF);
      const int n0 = col, n1 = col + 16;
      float bias0 = u0 * bb[n0] + u1 * bb[OUTF + n0]
                  + u2 * bb[2 * OUTF + n0] + u3 * bb[3 * OUTF + n0];
      float bias1 = u0 * bb[n1] + u1 * bb[OUTF + n1]
                  + u2 * bb[2 * OUTF + n1] + u3 * bb[3 * OUTF + n1];
      atomicAdd(out + d * OUTF + n0, acc0[v] + bias0);
      atomicAdd(out + d * OUTF + n1, acc1[v] + bias1);
    }
  }
}

// ---------------------------------------------------------------------------
extern "C" void kernel_launch(void* const* d_in, const int* in_sizes, int n_in,
                              void* d_out, int out_size, void* d_ws, size_t ws_size,
                              hipStream_t stream) {
  const float* feat = (const float*)d_in[0];   // (N, 32)
  const float* tv   = (const float*)d_in[1];   // (E, 1, 4)
  const float* wgt  = (const float*)d_in[2];   // (8, 4, 32, 32)
  const float* hb   = (const float*)d_in[3];   // (8, 4, 32)
  const int*   et   = (const int*)d_in[4];     // (E,)
  const int*   src  = (const int*)d_in[5];     // (E,)
  const int*   dst  = (const int*)d_in[6];     // (E,)
  float* out = (float*)d_out;                  // (N, 1, 32) flat
  const int E = in_sizes[4];

  // workspace layout: [counts:8][cursor:8][pad][edge_order:E]
  int* wsI    = (int*)d_ws;
  int* counts = wsI;
  int* cursor = wsI + NRELS;
  int* order  = wsI + 32;

  k_init   <<<(out_size + 255) / 256, 256, 0, stream>>>(out, out_size, counts);
  k_hist   <<<(E + 255) / 256, 256, 0, stream>>>(et, counts, E);
  k_scan   <<<1, 32, 0, stream>>>(counts, cursor);
  k_scatter<<<(E + 255) / 256, 256, 0, stream>>>(et, cursor, order, E);

  const int ntiles = (E + TILE - 1) / TILE;
  k_rgcn_wmma<<<ntiles, 32, 0, stream>>>(feat, tv, wgt, hb, et, src, dst,
                                         order, out, E);
}